// MultiModalSDTPS_25374666785594
// MI455X (gfx1250) — compile-verified
//
#include <hip/hip_runtime.h>
#include <math.h>

// Problem constants (B,N,C)=(128,512,512)
#define BB      128
#define NN      512
#define CC      512
#define HSP     128      // C/4
#define NKEEP   256      // ceil(N*0.5)
#define NCV     102      // int(N*0.4*0.5)
#define HAGV    102      // int(C*0.2)
#define PADW    112      // 102 padded to 7*16
#define ENH_ROWS 103     // 102 aggr rows + 1 extra row

typedef __attribute__((ext_vector_type(2))) float v2f;
typedef __attribute__((ext_vector_type(8))) float v8f;

// D = A(16x4) * B(4x16) + C  (fp32 WMMA, wave32)
__device__ __forceinline__ v8f wmma4(v2f a, v2f b, v8f c) {
  return __builtin_amdgcn_wmma_f32_16x16x4_f32(false, a, false, b, (short)0, c,
                                               false, false);
}

__device__ __forceinline__ float warp_sum(float v) {
#pragma unroll
  for (int m = 16; m >= 1; m >>= 1) v += __shfl_xor(v, m, 32);
  return v;
}
__device__ __forceinline__ float half_sum(float v) {   // reduce within 16-lane group
#pragma unroll
  for (int m = 8; m >= 1; m >>= 1) v += __shfl_xor(v, m, 32);
  return v;
}
__device__ __forceinline__ float warp_min(float v) {
#pragma unroll
  for (int m = 16; m >= 1; m >>= 1) v = fminf(v, __shfl_xor(v, m, 32));
  return v;
}
__device__ __forceinline__ float warp_max(float v) {
#pragma unroll
  for (int m = 16; m >= 1; m >>= 1) v = fmaxf(v, __shfl_xor(v, m, 32));
  return v;
}
__device__ __forceinline__ float gelu_exact(float x) {
  return 0.5f * x * (1.0f + erff(x * 0.70710678118654752f));
}

// ---------------------------------------------------------------------------
// Kernel 1: s_pred = sigmoid(gelu(tok @ sp_w1 + b1) @ sp_w2 + b2)
// One wave per 16-token tile; 8 accumulator tiles cover HID_SP=128.
// grid 512 x 256  ->  4096 waves == 65536/16 tiles
// ---------------------------------------------------------------------------
__global__ __launch_bounds__(256)
void spred_kernel(const float* __restrict__ tok, const float* __restrict__ w1,
                  const float* __restrict__ b1, const float* __restrict__ w2,
                  const float* __restrict__ b2, float* __restrict__ spred) {
  const int wave = (blockIdx.x * blockDim.x + threadIdx.x) >> 5;
  const int lane = threadIdx.x & 31;
  const int hf = lane >> 4, lo = lane & 15;
  const long rb = (long)wave * 16;

  v8f acc[8];
#pragma unroll
  for (int t = 0; t < 8; ++t) acc[t] = (v8f){0,0,0,0,0,0,0,0};

  for (int k = 0; k < CC; k += 4) {
    const int ka = k + 2 * hf;
    const float* ap = tok + (rb + lo) * CC + ka;
    v2f a; a.x = ap[0]; a.y = ap[1];
#pragma unroll
    for (int nt = 0; nt < 8; ++nt) {
      const int col = nt * 16 + lo;
      v2f bf;
      bf.x = w1[ka * HSP + col];
      bf.y = w1[(ka + 1) * HSP + col];
      acc[nt] = wmma4(a, bf, acc[nt]);
    }
  }
  float p[8];
#pragma unroll
  for (int r = 0; r < 8; ++r) p[r] = 0.f;
#pragma unroll
  for (int nt = 0; nt < 8; ++nt) {
    const int col = nt * 16 + lo;
    const float bb = b1[col], ww = w2[col];
#pragma unroll
    for (int r = 0; r < 8; ++r) {
      const float h = acc[nt][r] + bb;
      p[r] += gelu_exact(h) * ww;
    }
  }
#pragma unroll
  for (int r = 0; r < 8; ++r) p[r] = half_sum(p[r]);
  if (lo == 0) {
    const float bb2 = b2[0];
#pragma unroll
    for (int r = 0; r < 8; ++r) {
      const float v = p[r] + bb2;
      spred[rb + r + 8 * hf] = 1.0f / (1.0f + expf(-v));
    }
  }
}

// ---------------------------------------------------------------------------
// Kernel 2: cos sims, min-max, score, bitonic argsort, mask, keep idx,
// softmax over non-keep, "extra" row.  One block (256 thr) per batch.
// ---------------------------------------------------------------------------
__global__ __launch_bounds__(256)
void score_kernel(const float* __restrict__ tok, const float* __restrict__ gim,
                  const float* __restrict__ g2, const float* __restrict__ g3,
                  const float* __restrict__ spred, int* __restrict__ keep,
                  float* __restrict__ mask_out, float* __restrict__ enh_out) {
  const int b = blockIdx.x;
  const int tid = threadIdx.x;
  const int wave = tid >> 5, lane = tid & 31;

  __shared__ float sc1[NN], sc2[NN], sc3[NN], ss[NN];
  __shared__ int sidx[NN];
  __shared__ float sred[48];
  __shared__ float gnorm[3];
  __shared__ float mmv[6];
  __shared__ float rtmp[256];

  // global vector norms
  if (wave < 3) {
    const float* g = (wave == 0) ? gim : ((wave == 1) ? g2 : g3);
    float s = 0.f;
    for (int c = lane; c < CC; c += 32) { float v = g[b * CC + c]; s += v * v; }
    s = warp_sum(s);
    if (lane == 0) gnorm[wave] = fmaxf(sqrtf(s), 1e-12f);
  }
  __syncthreads();

  const float* tb = tok + (size_t)b * NN * CC;
  for (int t = wave; t < NN; t += 8) {
    const float* tr = tb + (size_t)t * CC;
    float d1 = 0, d2 = 0, d3 = 0, nn = 0;
    for (int c = lane; c < CC; c += 32) {
      const float x = tr[c];
      d1 += x * gim[b * CC + c];
      d2 += x * g2[b * CC + c];
      d3 += x * g3[b * CC + c];
      nn += x * x;
    }
    d1 = warp_sum(d1); d2 = warp_sum(d2); d3 = warp_sum(d3); nn = warp_sum(nn);
    if (lane == 0) {
      const float nt = fmaxf(sqrtf(nn), 1e-12f);
      sc1[t] = d1 / (nt * gnorm[0]);
      sc2[t] = d2 / (nt * gnorm[1]);
      sc3[t] = d3 / (nt * gnorm[2]);
    }
  }
  __syncthreads();

  // per-batch min/max of the three cos arrays
  float mn[3] = {1e30f, 1e30f, 1e30f}, mx[3] = {-1e30f, -1e30f, -1e30f};
  for (int i = tid; i < NN; i += 256) {
    mn[0] = fminf(mn[0], sc1[i]); mx[0] = fmaxf(mx[0], sc1[i]);
    mn[1] = fminf(mn[1], sc2[i]); mx[1] = fmaxf(mx[1], sc2[i]);
    mn[2] = fminf(mn[2], sc3[i]); mx[2] = fmaxf(mx[2], sc3[i]);
  }
#pragma unroll
  for (int q = 0; q < 3; ++q) { mn[q] = warp_min(mn[q]); mx[q] = warp_max(mx[q]); }
  if (lane == 0) {
#pragma unroll
    for (int q = 0; q < 3; ++q) {
      sred[(2 * q) * 8 + wave] = mn[q];
      sred[(2 * q + 1) * 8 + wave] = mx[q];
    }
  }
  __syncthreads();
  if (tid < 6) {
    float v = sred[tid * 8];
    for (int w = 1; w < 8; ++w)
      v = (tid & 1) ? fmaxf(v, sred[tid * 8 + w]) : fminf(v, sred[tid * 8 + w]);
    mmv[tid] = v;
  }
  __syncthreads();

  const float i1 = 1.f / (mmv[1] - mmv[0] + 1e-8f);
  const float i2 = 1.f / (mmv[3] - mmv[2] + 1e-8f);
  const float i3 = 1.f / (mmv[5] - mmv[4] + 1e-8f);
  for (int i = tid; i < NN; i += 256) {
    const float m1 = (sc1[i] - mmv[0]) * i1;
    const float m2 = (sc2[i] - mmv[2]) * i2;
    const float m3 = (sc3[i] - mmv[4]) * i3;
    ss[i] = 0.5f * spred[b * NN + i] + 0.25f * (m2 + m3 + 2.0f * m1);
    sidx[i] = i;
  }
  __syncthreads();

  // bitonic sort: descending, ties -> smaller index first (stable argsort(-s))
  for (int k = 2; k <= NN; k <<= 1) {
    for (int j = k >> 1; j > 0; j >>= 1) {
      for (int i = tid; i < NN; i += 256) {
        const int l = i ^ j;
        if (l > i) {
          const bool dirUp = ((i & k) == 0);
          const float si = ss[i], sl = ss[l];
          const int ii = sidx[i], il = sidx[l];
          const bool bef = (si > sl) || (si == sl && ii < il);
          if (dirUp ? !bef : bef) {
            ss[i] = sl; ss[l] = si; sidx[i] = il; sidx[l] = ii;
          }
        }
      }
      __syncthreads();
    }
  }

  // score_mask + keep indices
  for (int i = tid; i < NN; i += 256)
    mask_out[b * NN + sidx[i]] = (i < NKEEP) ? 1.0f : 0.0f;
  if (tid < NKEEP) keep[b * NKEEP + tid] = sidx[tid];

  // softmax over sorted non-keep scores (max is ss[NKEEP], sorted descending)
  const float smax = ss[NKEEP];
  const float e = expf(ss[NKEEP + tid] - smax);
  rtmp[tid] = e;
  sc1[tid] = e;          // weights stash (sc1 no longer needed)
  __syncthreads();
  for (int st = 128; st >= 1; st >>= 1) {
    if (tid < st) rtmp[tid] += rtmp[tid + st];
    __syncthreads();
  }
  const float winv = 1.0f / rtmp[0];
  __syncthreads();

  // extra = sum_j w_j * tokens[b, order[NKEEP+j], :]
  for (int c = tid; c < CC; c += 256) {
    float acc = 0.f;
    for (int j = 0; j < NKEEP; ++j)
      acc += sc1[j] * winv * tb[(size_t)sidx[NKEEP + j] * CC + c];
    enh_out[((size_t)b * ENH_ROWS + NCV) * CC + c] = acc;
  }
}

// ---------------------------------------------------------------------------
// Kernel 3: gather kept tokens + layernorm -> xn.  One wave per kept row.
// grid 4096 x 256 -> 32768 waves
// ---------------------------------------------------------------------------
__global__ __launch_bounds__(256)
void lnsel_kernel(const float* __restrict__ tok, const int* __restrict__ keep,
                  const float* __restrict__ g, const float* __restrict__ bt,
                  float* __restrict__ xn) {
  const int gw = (blockIdx.x * blockDim.x + threadIdx.x) >> 5;
  const int lane = threadIdx.x & 31;
  const int b = gw >> 8, s = gw & 255;
  const int src = keep[gw];
  const float* row = tok + ((size_t)b * NN + src) * CC;
  float sum = 0.f, sq = 0.f;
  for (int c = lane; c < CC; c += 32) { const float x = row[c]; sum += x; sq += x * x; }
  sum = warp_sum(sum); sq = warp_sum(sq);
  const float mu = sum * (1.0f / CC);
  const float var = sq * (1.0f / CC) - mu * mu;
  const float rs = rsqrtf(var + 1e-5f);
  float* orow = xn + (size_t)gw * CC;
  for (int c = lane; c < CC; c += 32)
    orow[c] = (row[c] - mu) * rs * g[c] + bt[c];
  (void)s;
}

// ---------------------------------------------------------------------------
// Kernel 4: h = gelu(xn @ ag_w1 + b1)   (32768x512)@(512x102) -> padded 112
// One wave per 16-row tile, 7 column accumulators. grid 256 x 256.
// ---------------------------------------------------------------------------
__global__ __launch_bounds__(256)
void ag1_kernel(const float* __restrict__ xn, const float* __restrict__ w1,
                const float* __restrict__ b1, float* __restrict__ h) {
  const int wave = (blockIdx.x * blockDim.x + threadIdx.x) >> 5;
  const int lane = threadIdx.x & 31;
  const int hf = lane >> 4, lo = lane & 15;
  const long rb = (long)wave * 16;

  v8f acc[7];
#pragma unroll
  for (int t = 0; t < 7; ++t) acc[t] = (v8f){0,0,0,0,0,0,0,0};

  for (int k = 0; k < CC; k += 4) {
    const int ka = k + 2 * hf;
    const float* ap = xn + (rb + lo) * CC + ka;
    v2f a; a.x = ap[0]; a.y = ap[1];
#pragma unroll
    for (int ct = 0; ct < 7; ++ct) {
      const int col = ct * 16 + lo;
      v2f bf;
      bf.x = (col < HAGV) ? w1[ka * HAGV + col] : 0.f;
      bf.y = (col < HAGV) ? w1[(ka + 1) * HAGV + col] : 0.f;
      acc[ct] = wmma4(a, bf, acc[ct]);
    }
  }
#pragma unroll
  for (int ct = 0; ct < 7; ++ct) {
    const int col = ct * 16 + lo;
    const float bb = (col < HAGV) ? b1[col] : 0.f;
#pragma unroll
    for (int r = 0; r < 8; ++r) {
      const long row = rb + r + 8 * hf;
      float v = 0.f;
      if (col < HAGV) v = gelu_exact(acc[ct][r] + bb);
      h[row * PADW + col] = v;   // zero pad cols so GEMM2 K-padding is safe
    }
  }
}

// ---------------------------------------------------------------------------
// Kernel 5: w_raw = h @ ag_w2 + b2   (32768x102)@(102x102), K padded to 112
// ---------------------------------------------------------------------------
__global__ __launch_bounds__(256)
void ag2_kernel(const float* __restrict__ h, const float* __restrict__ w2,
                const float* __restrict__ b2, float* __restrict__ wraw) {
  const int wave = (blockIdx.x * blockDim.x + threadIdx.x) >> 5;
  const int lane = threadIdx.x & 31;
  const int hf = lane >> 4, lo = lane & 15;
  const long rb = (long)wave * 16;

  v8f acc[7];
#pragma unroll
  for (int t = 0; t < 7; ++t) acc[t] = (v8f){0,0,0,0,0,0,0,0};

  for (int k = 0; k < PADW; k += 4) {
    const int ka = k + 2 * hf;
    const float* ap = h + (rb + lo) * PADW + ka;
    v2f a; a.x = ap[0]; a.y = ap[1];   // pad cols of h are zero
#pragma unroll
    for (int ct = 0; ct < 7; ++ct) {
      const int col = ct * 16 + lo;
      v2f bf;
      bf.x = (ka < HAGV && col < NCV) ? w2[ka * NCV + col] : 0.f;
      bf.y = (ka + 1 < HAGV && col < NCV) ? w2[(ka + 1) * NCV + col] : 0.f;
      acc[ct] = wmma4(a, bf, acc[ct]);
    }
  }
#pragma unroll
  for (int ct = 0; ct < 7; ++ct) {
    const int col = ct * 16 + lo;
    if (col < NCV) {
      const float bb = b2[col];
#pragma unroll
      for (int r = 0; r < 8; ++r) {
        const long row = rb + r + 8 * hf;
        wraw[row * PADW + col] = acc[ct][r] + bb;
      }
    }
  }
}

// ---------------------------------------------------------------------------
// Kernel 6: softmax over the 256 kept tokens per (b,nc); writes transposed
// wsoft[b][nc][s].  One wave per (b,nc). grid 1632 x 256.
// ---------------------------------------------------------------------------
__global__ __launch_bounds__(256)
void softw_kernel(const float* __restrict__ wraw, const float* __restrict__ scale,
                  float* __restrict__ wsoft) {
  const int gw = (blockIdx.x * blockDim.x + threadIdx.x) >> 5;
  if (gw >= BB * NCV) return;
  const int lane = threadIdx.x & 31;
  const int b = gw / NCV, nc = gw % NCV;
  const float sc = scale[0];
  const float* base = wraw + (size_t)b * NKEEP * PADW + nc;
  float v[8];
  float mx = -1e30f;
#pragma unroll
  for (int i = 0; i < 8; ++i) {
    v[i] = base[(size_t)(lane + i * 32) * PADW] * sc;
    mx = fmaxf(mx, v[i]);
  }
  mx = warp_max(mx);
  float sum = 0.f;
#pragma unroll
  for (int i = 0; i < 8; ++i) { v[i] = expf(v[i] - mx); sum += v[i]; }
  sum = warp_sum(sum);
  const float inv = 1.0f / sum;
  float* out = wsoft + ((size_t)b * PADW + nc) * NKEEP;
#pragma unroll
  for (int i = 0; i < 8; ++i) out[lane + i * 32] = v[i] * inv;
}

// ---------------------------------------------------------------------------
// Kernel 7: aggr[b,nc,c] = sum_s wsoft[b,nc,s] * tok[b,keep[s],c]  (WMMA)
// grid (128, 7, 4) x 256: block=8 waves, wave -> one 16x16 tile (mt, ntile).
// ---------------------------------------------------------------------------
__global__ __launch_bounds__(256)
void aggr_kernel(const float* __restrict__ tok, const int* __restrict__ keep,
                 const float* __restrict__ wsoft, float* __restrict__ enh) {
  const int b = blockIdx.x, mt = blockIdx.y, ng = blockIdx.z;
  const int wave = threadIdx.x >> 5, lane = threadIdx.x & 31;
  const int hf = lane >> 4, lo = lane & 15;
  __shared__ int skeep[NKEEP];
  skeep[threadIdx.x] = keep[b * NKEEP + threadIdx.x];
  __syncthreads();

  const int ntile = ng * 8 + wave;
  const int col = ntile * 16 + lo;
  const int mrow = mt * 16 + lo;
  const bool mOk = (mrow < NCV);
  const float* arow = wsoft + ((size_t)b * PADW + (mOk ? mrow : 0)) * NKEEP;
  const float* tb = tok + (size_t)b * NN * CC;

  v8f acc = (v8f){0,0,0,0,0,0,0,0};
  for (int k = 0; k < NKEEP; k += 4) {
    const int ka = k + 2 * hf;
    v2f a;
    a.x = mOk ? arow[ka] : 0.f;
    a.y = mOk ? arow[ka + 1] : 0.f;
    v2f bf;
    bf.x = tb[(size_t)skeep[ka] * CC + col];
    bf.y = tb[(size_t)skeep[ka + 1] * CC + col];
    acc = wmma4(a, bf, acc);
  }
#pragma unroll
  for (int r = 0; r < 8; ++r) {
    const int m = mt * 16 + r + 8 * hf;
    if (m < NCV)
      enh[((size_t)b * ENH_ROWS + m) * CC + col] = acc[r];
  }
}

// ---------------------------------------------------------------------------
extern "C" void kernel_launch(void* const* d_in, const int* in_sizes, int n_in,
                              void* d_out, int out_size, void* d_ws, size_t ws_size,
                              hipStream_t stream) {
  (void)in_sizes; (void)n_in; (void)out_size; (void)ws_size;
  const float* tok[3] = {(const float*)d_in[0], (const float*)d_in[1],
                         (const float*)d_in[2]};
  const float* gl[3] = {(const float*)d_in[3], (const float*)d_in[4],
                        (const float*)d_in[5]};
  const int g2i[3] = {1, 0, 0};
  const int g3i[3] = {2, 2, 1};

  float* out = (float*)d_out;
  float* ws = (float*)d_ws;
  // workspace layout (floats): spred | keep(int) | xn (reused: wraw, wsoft) | h
  float* spred = ws;                                  // 65536
  int* keep = (int*)(ws + 65536);                     // 32768 ints
  float* xn = ws + 98304;                             // 16,777,216
  float* wraw = xn;                                   // 3,670,016 (xn dead)
  float* wsoft = xn + (size_t)(4u << 20);             // 3,670,016
  float* hag = ws + 98304 + (size_t)16777216;         // 3,670,016

  const size_t enh_sz = (size_t)BB * ENH_ROWS * CC;   // 6,750,208
  const size_t mask_sz = (size_t)BB * NN;             // 65,536

  for (int m = 0; m < 3; ++m) {
    const int pb = 6 + m * 11;
    const float* sp_w1 = (const float*)d_in[pb + 0];
    const float* sp_b1 = (const float*)d_in[pb + 1];
    const float* sp_w2 = (const float*)d_in[pb + 2];
    const float* sp_b2 = (const float*)d_in[pb + 3];
    const float* ln_g  = (const float*)d_in[pb + 4];
    const float* ln_b  = (const float*)d_in[pb + 5];
    const float* ag_w1 = (const float*)d_in[pb + 6];
    const float* ag_b1 = (const float*)d_in[pb + 7];
    const float* ag_w2 = (const float*)d_in[pb + 8];
    const float* ag_b2 = (const float*)d_in[pb + 9];
    const float* scale = (const float*)d_in[pb + 10];

    float* enh = out + (size_t)m * enh_sz;
    float* mask = out + 3 * enh_sz + (size_t)m * mask_sz;

    spred_kernel<<<512, 256, 0, stream>>>(tok[m], sp_w1, sp_b1, sp_w2, sp_b2, spred);
    score_kernel<<<BB, 256, 0, stream>>>(tok[m], gl[m], gl[g2i[m]], gl[g3i[m]],
                                         spred, keep, mask, enh);
    lnsel_kernel<<<4096, 256, 0, stream>>>(tok[m], keep, ln_g, ln_b, xn);
    ag1_kernel<<<256, 256, 0, stream>>>(xn, ag_w1, ag_b1, hag);
    ag2_kernel<<<256, 256, 0, stream>>>(hag, ag_w2, ag_b2, wraw);
    softw_kernel<<<1632, 256, 0, stream>>>(wraw, scale, wsoft);
    aggr_kernel<<<dim3(BB, 7, 4), 256, 0, stream>>>(tok[m], keep, wsoft, enh);
  }
}